// ZImageImage2LoRAModelCompressed_50766513438777
// MI455X (gfx1250) — compile-verified
//
#include <hip/hip_runtime.h>

#define EMB  1536
#define RANK 32
#define N3   204
#define N10  34
#define D3   3840
#define D10  10240

typedef __attribute__((ext_vector_type(2))) float v2f;
typedef __attribute__((ext_vector_type(8))) float v8f;

// ---------------------------------------------------------------------------
// Kernel 1: fused GEMV  P[n,k] = dot(Wp[n,k,:], x) + bp[n,k]  for both groups.
// One wave32 per 1536-long dot product; x staged in LDS; Wp streamed with
// coalesced float4 loads (lane-consecutive 16B chunks).
// ---------------------------------------------------------------------------
__global__ __launch_bounds__(256)
void gemv_proj_kernel(const float* __restrict__ x,
                      const float* __restrict__ Wp3,  const float* __restrict__ bp3,
                      const float* __restrict__ Wp10, const float* __restrict__ bp10,
                      float* __restrict__ Pout /* [(N3+N10)*1024] */) {
    __shared__ __align__(16) float xs[EMB];
    for (int i = threadIdx.x; i < EMB; i += 256) xs[i] = x[i];
    __syncthreads();

    const int wave = threadIdx.x >> 5;
    const int lane = threadIdx.x & 31;
    const int row  = blockIdx.x * 8 + wave;          // global dot index
    const int R3   = N3 * RANK * RANK;               // 208896

    const float* W; const float* b; int r;
    if (row < R3) { W = Wp3;  b = bp3;  r = row; }
    else          { W = Wp10; b = bp10; r = row - R3; }

    const float4* __restrict__ W4 = (const float4*)(W + (long long)r * EMB);
    const float4* __restrict__ x4 = (const float4*)xs;

    float s = 0.0f;
#pragma unroll
    for (int i = 0; i < EMB / 128; ++i) {            // 12 iterations
        float4 w  = W4[i * 32 + lane];
        float4 xv = x4[i * 32 + lane];
        s += w.x * xv.x + w.y * xv.y + w.z * xv.z + w.w * xv.w;
    }
#pragma unroll
    for (int off = 16; off > 0; off >>= 1) s += __shfl_xor(s, off, 32);
    if (lane == 0) Pout[row] = s + b[r];
}

// ---------------------------------------------------------------------------
// Kernel 2: la[n] = P[n] (32x32) @ A[n] (32xD) via V_WMMA_F32_16X16X4_F32.
// Block = 4 waves; block handles one module and 4 adjacent 16-column tiles.
// P lives in LDS (stride-33 padding), A is streamed from global as B-frags.
// Per wave: 2 M-halves x 8 K-steps = 16 chained WMMAs.
// ---------------------------------------------------------------------------
__global__ __launch_bounds__(128)
void lora_pa_wmma_kernel(const float* __restrict__ P,   // [nmod, 32, 32]
                         const float* __restrict__ A,   // [nmod, 32, D]
                         float* __restrict__ out,       // [nmod, 32, D]
                         int D) {
    __shared__ float Ps[32][33];

    const int n = blockIdx.y;
    const float* __restrict__ Pn = P + n * RANK * RANK;
    for (int i = threadIdx.x; i < RANK * RANK; i += 128)
        Ps[i >> 5][i & 31] = Pn[i];
    __syncthreads();

    const int wave   = threadIdx.x >> 5;
    const int lane   = threadIdx.x & 31;
    const int lhalf  = lane >> 4;                 // 0: lanes 0-15, 1: lanes 16-31
    const int l15    = lane & 15;
    const int col0   = (blockIdx.x * 4 + wave) * 16;

    const float* __restrict__ An = A   + (long long)n * RANK * D;
    float*       __restrict__ On = out + (long long)n * RANK * D;

#pragma unroll
    for (int m = 0; m < 2; ++m) {                 // two 16-row halves of the 32-row output
        v8f c = {0.f, 0.f, 0.f, 0.f, 0.f, 0.f, 0.f, 0.f};
#pragma unroll
        for (int k = 0; k < 8; ++k) {             // K = 32 in steps of 4
            const int kb = k * 4 + lhalf * 2;     // lanes 0-15: K={kb,kb+1}=({0,1}); 16-31: {2,3}
            v2f a, bf;
            a.x  = Ps[m * 16 + l15][kb + 0];      // A-frag: M = lane&15
            a.y  = Ps[m * 16 + l15][kb + 1];
            bf.x = An[(long long)(kb + 0) * D + col0 + l15];   // B-frag: N = lane&15
            bf.y = An[(long long)(kb + 1) * D + col0 + l15];
            c = __builtin_amdgcn_wmma_f32_16x16x4_f32(
                    /*neg_a=*/false, a, /*neg_b=*/false, bf,
                    /*c_mod=*/(short)0, c, /*reuse_a=*/false, /*reuse_b=*/false);
        }
        const int rbase = m * 16 + lhalf * 8;     // C layout: VGPR r -> M = rbase + r
#pragma unroll
        for (int r = 0; r < 8; ++r)
            On[(long long)(rbase + r) * D + col0 + l15] = c[r];
    }
}

// ---------------------------------------------------------------------------
extern "C" void kernel_launch(void* const* d_in, const int* in_sizes, int n_in,
                              void* d_out, int out_size, void* d_ws, size_t ws_size,
                              hipStream_t stream) {
    const float* x    = (const float*)d_in[0];
    const float* Wp3  = (const float*)d_in[1];
    const float* bp3  = (const float*)d_in[2];
    const float* Wp10 = (const float*)d_in[3];
    const float* bp10 = (const float*)d_in[4];
    const float* A3   = (const float*)d_in[5];
    const float* A10  = (const float*)d_in[6];
    const float* B3   = (const float*)d_in[7];
    const float* B10  = (const float*)d_in[8];
    float* out = (float*)d_out;

    float* Pbuf = (float*)d_ws;   // (N3+N10)*1024 floats = ~975 KB scratch

    const long long la3_elems  = (long long)N3  * RANK * D3;    // 25,067,520
    const long long la10_elems = (long long)N10 * RANK * D10;   // 11,141,120
    float* la3  = out;
    float* la10 = out + la3_elems;
    float* oB3  = la10 + la10_elems;
    float* oB10 = oB3 + la3_elems;      // B3 has same elem count as la3

    // B pass-through: pure DMA, no SIMD involvement.
    hipMemcpyAsync(oB3,  B3,  (size_t)la3_elems  * sizeof(float),
                   hipMemcpyDeviceToDevice, stream);
    hipMemcpyAsync(oB10, B10, (size_t)la10_elems * sizeof(float),
                   hipMemcpyDeviceToDevice, stream);

    // Stage 1: GEMV for both groups -> Pbuf.
    const int totalRows = (N3 + N10) * RANK * RANK;  // 243,712 (divisible by 8)
    gemv_proj_kernel<<<totalRows / 8, 256, 0, stream>>>(x, Wp3, bp3, Wp10, bp10, Pbuf);

    // Stage 2: WMMA P@A per group (stream-ordered after stage 1).
    dim3 g3 (D3  / 64, N3);
    dim3 g10(D10 / 64, N10);
    lora_pa_wmma_kernel<<<g3,  128, 0, stream>>>(Pbuf,                  A3,  la3,  D3);
    lora_pa_wmma_kernel<<<g10, 128, 0, stream>>>(Pbuf + N3 * RANK * RANK, A10, la10, D10);
}